// FLEigh_61435212202178
// MI455X (gfx1250) — compile-verified
//
#include <hip/hip_runtime.h>
#include <math.h>

typedef float v2f __attribute__((ext_vector_type(2)));
typedef float v8f __attribute__((ext_vector_type(8)));

#define LAG_ITERS 5
#define POLISH_ITERS 3
#define TA 32   // kernel A threads/block (one wave32)
#define WB 8    // kernel B waves/block

// ---------------------------------------------------------------------------
// Laguerre iteration + synthetic-division deflation, degree known at compile
// time so the coefficient array stays fully register-resident (SROA).
// ---------------------------------------------------------------------------
template<int DEG>
__device__ __forceinline__ double lag_deflate(double (&cl)[9], double z)
{
  #pragma unroll
  for (int it = 0; it < LAG_ITERS; ++it) {
    double pv = cl[DEG], dp = 0.0, d2 = 0.0;
    #pragma unroll
    for (int j = DEG - 1; j >= 0; --j) {
      d2 = fma(d2, z, dp);
      dp = fma(dp, z, pv);
      pv = fma(pv, z, cl[j]);
    }
    bool ok   = fabs(pv) > 1e-30;
    double ps = ok ? pv : 1.0;
    double G  = ok ? dp / ps : 0.0;
    double H  = fma(G, G, -(ok ? 2.0 * d2 / ps : 0.0));
    double disc = fmax((double)(DEG - 1) * ((double)DEG * H - G * G), 0.0);
    double sq = sqrt(disc);
    double gp = G + sq, gm = G - sq;
    double den = (fabs(gp) >= fabs(gm)) ? gp : gm;
    bool dok = fabs(den) > 1e-20;
    z = z - (dok ? (double)DEG / den : 0.0);
  }
  // deflate
  double b = cl[DEG];
  #pragma unroll
  for (int j = DEG - 1; j > 0; --j) { double bn = fma(z, b, cl[j]); cl[j] = b; b = bn; }
  cl[0] = b;
  return z;
}

// ---------------------------------------------------------------------------
// Kernel A: one matrix per thread. Faddeev-LeVerrier (f64, column-wise
// in-place), Laguerre roots + polish (f64), adjugate eigenvectors (f32 Horner
// over LDS-resident Ms). Writes un-orthogonalized V (row-major f32) to ws.
// ---------------------------------------------------------------------------
__global__ __launch_bounds__(TA) void eig_scalar_kernel(
    const float* __restrict__ A, float* __restrict__ Vws, int nMat)
{
  __shared__ float  msL[7 * 64 * TA];   // Ms[1..7] as f32, [kk][entry][tid]
  __shared__ double csL[9 * TA];        // char-poly coefficients
  __shared__ double rtL[8 * TA];        // polished roots

  const int tid = threadIdx.x;
  const int mat = blockIdx.x * TA + tid;
  if (mat >= nMat) return;
  const float* Ap = A + (size_t)mat * 64;

  float Af[64];
  #pragma unroll
  for (int i = 0; i < 16; ++i) {
    float4 v = reinterpret_cast<const float4*>(Ap)[i];
    Af[4*i+0] = v.x; Af[4*i+1] = v.y; Af[4*i+2] = v.z; Af[4*i+3] = v.w;
  }
  float ss = 0.f;
  #pragma unroll
  for (int e = 0; e < 64; ++e) ss = fmaf(Af[e], Af[e], ss);
  float scale = fmaxf(sqrtf(ss) * 0.35355339059327373f, 1e-12f); // /sqrt(8)
  float invs  = 1.0f / scale;

  double Ad[64];
  #pragma unroll
  for (int e = 0; e < 64; ++e) Ad[e] = (double)(Af[e] * invs);

  double dd[8];
  #pragma unroll
  for (int i = 0; i < 8; ++i) dd[i] = (double)(Af[i*9] * invs);

  // ---- Faddeev-LeVerrier: M <- Ad*M + ck*I, column-wise in place ----
  double M[64];
  #pragma unroll
  for (int e = 0; e < 64; ++e) M[e] = 0.0;
  double ck = 1.0;
  csL[8*TA + tid] = 1.0;
  for (int k = 1; k <= 8; ++k) {
    double tr = 0.0;
    #pragma unroll
    for (int j = 0; j < 8; ++j) {
      double t[8];
      #pragma unroll
      for (int i = 0; i < 8; ++i) {
        double s = 0.0;
        #pragma unroll
        for (int q = 0; q < 8; ++q) s = fma(Ad[i*8+q], M[q*8+j], s);
        t[i] = s;
      }
      #pragma unroll
      for (int i = 0; i < 8; ++i) {
        double v = t[i] + ((i == j) ? ck : 0.0);
        M[i*8+j] = v;
        tr = fma(Ad[i*8+j], v, tr);
      }
    }
    if (k >= 2) {
      const int kk = k - 2;
      #pragma unroll
      for (int e = 0; e < 64; ++e) msL[(kk*64 + e)*TA + tid] = (float)M[e];
    }
    ck = -tr / (double)k;
    csL[(8-k)*TA + tid] = ck;
  }

  double c[9], cl[9];
  #pragma unroll
  for (int j = 0; j < 9; ++j) { c[j] = csL[j*TA + tid]; cl[j] = c[j]; }

  // sorted diagonal as initial guesses (Batcher-8 network on min/max)
  #define CSWD(a,b) { double lo_ = fmin(a,b), hi_ = fmax(a,b); a = lo_; b = hi_; }
  CSWD(dd[0],dd[1]); CSWD(dd[2],dd[3]); CSWD(dd[4],dd[5]); CSWD(dd[6],dd[7]);
  CSWD(dd[0],dd[2]); CSWD(dd[1],dd[3]); CSWD(dd[4],dd[6]); CSWD(dd[5],dd[7]);
  CSWD(dd[1],dd[2]); CSWD(dd[5],dd[6]);
  CSWD(dd[0],dd[4]); CSWD(dd[1],dd[5]); CSWD(dd[2],dd[6]); CSWD(dd[3],dd[7]);
  CSWD(dd[2],dd[4]); CSWD(dd[3],dd[5]);
  CSWD(dd[1],dd[2]); CSWD(dd[3],dd[4]); CSWD(dd[5],dd[6]);
  #undef CSWD

  const double st = 2e-4 / 7.0;
  double roots[8];
  roots[0] = lag_deflate<8>(cl, dd[0] - 1e-4 + 0.0*st);
  roots[1] = lag_deflate<7>(cl, dd[1] - 1e-4 + 1.0*st);
  roots[2] = lag_deflate<6>(cl, dd[2] - 1e-4 + 2.0*st);
  roots[3] = lag_deflate<5>(cl, dd[3] - 1e-4 + 3.0*st);
  roots[4] = lag_deflate<4>(cl, dd[4] - 1e-4 + 4.0*st);
  roots[5] = lag_deflate<3>(cl, dd[5] - 1e-4 + 5.0*st);
  roots[6] = lag_deflate<2>(cl, dd[6] - 1e-4 + 6.0*st);
  roots[7] = lag_deflate<1>(cl, dd[7] - 1e-4 + 7.0*st);

  // Newton polish on the full (monic) polynomial
  for (int p = 0; p < POLISH_ITERS; ++p) {
    #pragma unroll
    for (int r = 0; r < 8; ++r) {
      double z = roots[r], pv = 1.0, dp = 0.0;
      #pragma unroll
      for (int j = 7; j >= 0; --j) { dp = fma(dp, z, pv); pv = fma(pv, z, c[j]); }
      bool ok = fabs(dp) > 1e-30;
      roots[r] = z - (ok ? pv / dp : 0.0);
    }
  }
  #pragma unroll
  for (int r = 0; r < 8; ++r) rtL[r*TA + tid] = roots[r];

  // ---- adjugate eigenvectors: R(lam) = Horner over {I, Ms[1..7]} ----
  float* Vp = Vws + (size_t)mat * 64;
  for (int ri = 0; ri < 8; ++ri) {
    const float lam = (float)rtL[ri*TA + tid];
    float nrm[8];
    #pragma unroll
    for (int j = 0; j < 8; ++j) nrm[j] = 0.f;
    for (int i = 0; i < 8; ++i) {
      #pragma unroll
      for (int j = 0; j < 8; ++j) {
        float r = (i == j) ? 1.0f : 0.0f;
        #pragma unroll
        for (int kk = 0; kk < 7; ++kk)
          r = fmaf(r, lam, msL[(kk*64 + i*8 + j)*TA + tid]);
        nrm[j] = fmaf(r, r, nrm[j]);
      }
    }
    int best = 0; float bb = nrm[0];
    #pragma unroll
    for (int j = 1; j < 8; ++j) { if (nrm[j] > bb) { bb = nrm[j]; best = j; } }
    float s2 = 0.f;
    for (int i = 0; i < 8; ++i) {
      float r = (i == best) ? 1.0f : 0.0f;
      #pragma unroll
      for (int kk = 0; kk < 7; ++kk)
        r = fmaf(r, lam, msL[(kk*64 + i*8 + best)*TA + tid]);
      s2 = fmaf(r, r, s2);
    }
    const float inv = 1.0f / (sqrtf(s2) + 1e-30f);
    for (int i = 0; i < 8; ++i) {
      float r = (i == best) ? 1.0f : 0.0f;
      #pragma unroll
      for (int kk = 0; kk < 7; ++kk)
        r = fmaf(r, lam, msL[(kk*64 + i*8 + best)*TA + tid]);
      Vp[i*8 + ri] = r * inv;   // V[:, ri] = normalized column
    }
  }
}

// ---------------------------------------------------------------------------
// Kernel B: WMMA Newton-Schulz. Two 8x8 matrices packed block-diagonally in a
// 16x16 tile per wave; one 16x16x16 product = 4 x V_WMMA_F32_16X16X4_F32.
// ---------------------------------------------------------------------------
__device__ __forceinline__ void mm16(const float* Asrc, bool transA,
                                     const float* Bsrc, float* Dst,
                                     int lane, float scl)
{
  const int m = lane & 15, half = lane >> 4;
  v8f acc = {};
  #pragma unroll
  for (int ks = 0; ks < 4; ++ks) {
    const int kb = ks * 4 + 2 * half;      // A-layout: K = vgpr + 2*(lane>>4)
    v2f a, b;
    if (transA) { a.x = Asrc[(kb+0)*16 + m]; a.y = Asrc[(kb+1)*16 + m]; }
    else        { a.x = Asrc[m*16 + kb+0];   a.y = Asrc[m*16 + kb+1];   }
    b.x = Bsrc[(kb+0)*16 + m];             // B-layout mirrors A (K rows)
    b.y = Bsrc[(kb+1)*16 + m];
    acc = __builtin_amdgcn_wmma_f32_16x16x4_f32(false, a, false, b,
                                                (short)0, acc, false, false);
  }
  #pragma unroll
  for (int r = 0; r < 8; ++r)              // C/D layout: M = r + 8*(lane>>4)
    Dst[(r + 8*half)*16 + m] = acc[r] * scl;
}

__global__ __launch_bounds__(WB*32) void ns_wmma_kernel(
    const float* __restrict__ A, const float* __restrict__ Vws,
    float* __restrict__ se, float* __restrict__ sv, int nMat)
{
  __shared__ float tile[WB][5][256];
  __shared__ float evL[WB][16];
  __shared__ int   pmL[WB][16];

  const int lane = threadIdx.x & 31;
  const int w    = threadIdx.x >> 5;
  const int pair = blockIdx.x * WB + w;
  const int m0   = pair * 2;
  if (m0 >= nMat) return;
  const bool has1 = (m0 + 1) < nMat;

  float* TV = &tile[w][0][0];
  float* TA_ = &tile[w][1][0];
  float* TY = &tile[w][2][0];
  float* TT = &tile[w][3][0];
  float* TX = &tile[w][4][0];

  const float* V0 = Vws + (size_t)m0 * 64;
  const float* V1 = Vws + (size_t)(m0+1) * 64;
  const float* A0 = A + (size_t)m0 * 64;
  const float* A1 = A + (size_t)(m0+1) * 64;

  #pragma unroll
  for (int r = 0; r < 8; ++r) {            // block-diagonal packing
    int idx = lane + 32*r, i = idx >> 4, j = idx & 15;
    float vv = 0.f, aa = 0.f;
    if (i < 8 && j < 8)                    { vv = V0[i*8+j];          aa = A0[i*8+j]; }
    else if (i >= 8 && j >= 8 && has1)     { vv = V1[(i-8)*8+(j-8)];  aa = A1[(i-8)*8+(j-8)]; }
    TV[idx] = vv; TA_[idx] = aa;
  }
  asm volatile("s_wait_dscnt 0x0" ::: "memory");

  mm16(TV, true,  TV, TY, lane, 1.0f);     // Y0 = V^T V
  #pragma unroll
  for (int r = 0; r < 8; ++r) {            // T1 = 3I - Y0 ; X1 = 0.5*T1
    int idx = lane + 32*r, i = idx >> 4, j = idx & 15;
    float t = ((i == j) ? 3.0f : 0.0f) - TY[idx];
    TT[idx] = t; TX[idx] = 0.5f * t;
  }
  mm16(TT, false, TY, TY, lane, 0.5f);     // Y1 = 0.5*T1*Y0
  #pragma unroll
  for (int r = 0; r < 8; ++r) {            // T2 = 3I - Y1
    int idx = lane + 32*r, i = idx >> 4, j = idx & 15;
    TT[idx] = ((i == j) ? 3.0f : 0.0f) - TY[idx];
  }
  mm16(TX, false, TT, TX, lane, 0.5f);     // X2 = 0.5*X1*T2
  mm16(TV, false, TX, TV, lane, 1.0f);     // V' = V*X2
  mm16(TA_, false, TV, TT, lane, 1.0f);    // AV = A*V'
  asm volatile("s_wait_dscnt 0x0" ::: "memory");

  if (lane < 16) {                         // evals: column dots
    float s = 0.f;
    #pragma unroll
    for (int i = 0; i < 16; ++i) s = fmaf(TV[i*16+lane], TT[i*16+lane], s);
    evL[w][lane] = s;
  }
  asm volatile("s_wait_dscnt 0x0" ::: "memory");

  if (lane < 2 && (lane == 0 || has1)) {   // per-matrix argsort (Batcher-8)
    const int h = lane;
    float v0 = evL[w][h*8+0], v1 = evL[w][h*8+1], v2 = evL[w][h*8+2], v3 = evL[w][h*8+3];
    float v4 = evL[w][h*8+4], v5 = evL[w][h*8+5], v6 = evL[w][h*8+6], v7 = evL[w][h*8+7];
    int i0=0,i1=1,i2=2,i3=3,i4=4,i5=5,i6=6,i7=7;
    #define CSW(va,vb,ia,ib) { if (va > vb) { float tf_=va; va=vb; vb=tf_; int ti_=ia; ia=ib; ib=ti_; } }
    CSW(v0,v1,i0,i1); CSW(v2,v3,i2,i3); CSW(v4,v5,i4,i5); CSW(v6,v7,i6,i7);
    CSW(v0,v2,i0,i2); CSW(v1,v3,i1,i3); CSW(v4,v6,i4,i6); CSW(v5,v7,i5,i7);
    CSW(v1,v2,i1,i2); CSW(v5,v6,i5,i6);
    CSW(v0,v4,i0,i4); CSW(v1,v5,i1,i5); CSW(v2,v6,i2,i6); CSW(v3,v7,i3,i7);
    CSW(v2,v4,i2,i4); CSW(v3,v5,i3,i5);
    CSW(v1,v2,i1,i2); CSW(v3,v4,i3,i4); CSW(v5,v6,i5,i6);
    #undef CSW
    float* sep = se + (size_t)(m0 + h) * 8;
    sep[0]=v0; sep[1]=v1; sep[2]=v2; sep[3]=v3;
    sep[4]=v4; sep[5]=v5; sep[6]=v6; sep[7]=v7;
    pmL[w][h*8+0]=i0; pmL[w][h*8+1]=i1; pmL[w][h*8+2]=i2; pmL[w][h*8+3]=i3;
    pmL[w][h*8+4]=i4; pmL[w][h*8+5]=i5; pmL[w][h*8+6]=i6; pmL[w][h*8+7]=i7;
  }
  asm volatile("s_wait_dscnt 0x0" ::: "memory");

  #pragma unroll
  for (int e = 0; e < 4; ++e) {            // cooperative permuted writeback
    int f = lane * 4 + e;                  // 0..127 = 2 matrices * 64
    int h = f >> 6;
    if (h == 1 && !has1) continue;
    int rem = f & 63, i = rem >> 3, r = rem & 7;
    int col = pmL[w][h*8 + r];
    sv[(size_t)(m0 + h) * 64 + i*8 + r] = TV[(h*8 + i)*16 + (h*8 + col)];
  }
}

// ---------------------------------------------------------------------------
extern "C" void kernel_launch(void* const* d_in, const int* in_sizes, int n_in,
                              void* d_out, int out_size, void* d_ws, size_t ws_size,
                              hipStream_t stream)
{
  (void)n_in; (void)out_size; (void)ws_size;
  const float* A = (const float*)d_in[0];
  const int nMat = in_sizes[0] / 64;

  float* Vws = (float*)d_ws;                       // nMat*64 f32 scratch
  float* se  = (float*)d_out;                      // (B,8) eigenvalues
  float* sv  = se + (size_t)nMat * 8;              // (B,8,8) eigenvectors

  dim3 gA((nMat + TA - 1) / TA), bA(TA);
  eig_scalar_kernel<<<gA, bA, 0, stream>>>(A, Vws, nMat);

  const int pairs = (nMat + 1) / 2;
  dim3 gB((pairs + WB - 1) / WB), bB(WB * 32);
  ns_wmma_kernel<<<gB, bB, 0, stream>>>(A, Vws, se, sv, nMat);
}